// RenderLoop_46145128628272
// MI455X (gfx1250) — compile-verified
//
#include <hip/hip_runtime.h>
#include <math.h>

// ---------------------------------------------------------------------------
// Fused ray-march renderer for MI455X (gfx1250, wave32).
//
// Per-pixel fp32 pipeline. Roofline: stores ~61MB => ~2.6us at 23.3 TB/s, so
// the kernel is VALU/transcendental bound; 20000 waves give ample latency
// hiding. Hot path is the sphere-march loop (~33 sqrt per leg): all
// sqrt/rcp/rsq/exp use raw hardware transcendentals (v_sqrt_f32 etc., ~1 ULP).
// The 6 per-leg cosines are folded into ONE sincosf via angle-addition and
// double-angle identities (phases are 0, 2pi/3, 4pi/3 and spin angle = 2x
// tang angle exactly).
//
// CDNA5-specific paths used:
//   * V_WMMA_F32_16X16X4_F32 for the tetrahedral-normal reduction
//       D(16x16) = A(16x4: [OFFS^T ; ones ; 0]) x B(4x16: d_k per pixel col)
//     -> rows 0..2 of D = gradient g, row 3 = sum(d_k) (Laplacian term),
//        rows 4..15 = 0 (exploited by the branchless merge below).
//   * ds_bpermute_b32 (full EXEC, unconditional) for data marshalling.
//
// Layout (per CDNA5 ISA):
//   A 16x4 f32 : lane L, vgpr v -> M = L&15, K = 2*(L>=16) + v
//   B 4x16 f32 : lane L, vgpr v -> N = L&15, K = 2*(L>=16) + v   (mirrors A)
//   D 16x16 f32: vgpr r -> lanes 0-15: M=r, N=lane; lanes 16-31: M=r+8 (=0 here)
//
// Merge trick: merged[j] = D0[j] + bpermute(D1[j], lane^16).
//   lane>=16 gathers D1[j][lane-16] (valid) and adds D0[j][lane] = row j+8 = 0;
//   lane<16  gathers D1[j][lane+16] = row j+8 = 0 and keeps D0[j][lane].
// Both operands are consumed by every lane, so the bpermute executes with
// full EXEC (bpermute reads from disabled source lanes return zero).
//
// EXEC is all-ones at every WMMA (grid exactly 2500x256, uniform control
// flow), as the ISA requires.
// ---------------------------------------------------------------------------

#define PXN     800
#define FOCAL_C 0.017f
#define SENS_C  0.017f
#define NEPS_C  0.05f
#define DECAY_C 0.01f

typedef __attribute__((ext_vector_type(2))) float v2f;
typedef __attribute__((ext_vector_type(8))) float v8f;

// Raw hardware transcendentals (~1 ULP) — skip IEEE libcall fixup sequences.
__device__ __forceinline__ float hsqrt(float x) { return __builtin_amdgcn_sqrtf(x); }
__device__ __forceinline__ float hrcp (float x) { return __builtin_amdgcn_rcpf(x); }
__device__ __forceinline__ float hrsq (float x) { return __builtin_amdgcn_rsqf(x); }
__device__ __forceinline__ float hexp (float x) {
    return __builtin_amdgcn_exp2f(x * 1.4426950408889634f);   // exp(x) = 2^(x*log2e)
}

__device__ __forceinline__ float sdf3(float x, float y, float z) {
    // min(|p| - 1, p.y + 1.5)
    return fminf(hsqrt(x * x + y * y + z * z) - 1.0f, y + 1.5f);
}

// OFFS rows (already scaled by NORMALS_EPS):
// k=0:(+,-,-)  k=1:(-,+,-)  k=2:(-,-,+)  k=3:(+,+,+)
__device__ __forceinline__ float offx(int k) { return (k == 0 || k == 3) ? NEPS_C : -NEPS_C; }
__device__ __forceinline__ float offy(int k) { return (k == 1 || k == 3) ? NEPS_C : -NEPS_C; }
__device__ __forceinline__ float offz(int k) { return (k == 2 || k == 3) ? NEPS_C : -NEPS_C; }

__device__ __forceinline__ float bperm_f(float v, int src_lane) {
    return __int_as_float(
        __builtin_amdgcn_ds_bpermute(src_lane << 2, __float_as_int(v)));
}

__global__ __launch_bounds__(256) void render_raymarch(
    const float* __restrict__ qin,   // (1,4) quaternion
    const float* __restrict__ tin,   // (1,3) translation
    const int*   __restrict__ dgp,   // degree
    const int*   __restrict__ stp,   // marching_steps
    const int*   __restrict__ lgp,   // legs
    float*       __restrict__ out)   // (8,1,800,800,3)
{
    const int gid  = blockIdx.x * blockDim.x + threadIdx.x;   // pixel id, exact fit
    const int w    = gid % PXN;
    const int h    = gid / PXN;
    const int lane = threadIdx.x & 31;
    const int nn   = lane & 15;          // WMMA column / row index
    const int k0   = (lane >> 4) * 2;    // this lane's K pair: {0,1} or {2,3}
    const int xidx = lane ^ 16;          // cross-half gather index

    // ---- uniform scalar state -------------------------------------------
    float qw = qin[0], qx = qin[1], qy = qin[2], qz = qin[3];
    {
        float r = hrsq(qw * qw + qx * qx + qy * qy + qz * qz);
        qw *= r; qx *= r; qy *= r; qz *= r;
    }
    float R00 = 1.f - 2.f * (qy * qy + qz * qz);
    float R01 = 2.f * (qx * qy - qw * qz);
    float R02 = 2.f * (qx * qz + qw * qy);
    float R10 = 2.f * (qx * qy + qw * qz);
    float R11 = 1.f - 2.f * (qx * qx + qz * qz);
    float R12 = 2.f * (qy * qz - qw * qx);
    float R20 = 2.f * (qx * qz - qw * qy);
    float R21 = 2.f * (qy * qz + qw * qx);
    float R22 = 1.f - 2.f * (qx * qx + qy * qy);

    const float tx = tin[0], ty = tin[1], tz = tin[2];
    const int   degree = dgp[0];
    const int   steps  = stp[0];
    const int   legs   = lgp[0];
    const float degf   = (float)degree;

    // ---- camera ray for this pixel --------------------------------------
    float sx = (((float)w + 0.5f) * (1.0f / (float)PXN) - 0.5f) * SENS_C;
    float sy = (((float)h + 0.5f) * (1.0f / (float)PXN) - 0.5f) * SENS_C;
    float sinv = hrsq(sx * sx + sy * sy + FOCAL_C * FOCAL_C);
    float dcx = sx * sinv, dcy = sy * sinv, dcz = FOCAL_C * sinv;
    float dx = R00 * dcx + R01 * dcy + R02 * dcz;
    float dy = R10 * dcx + R11 * dcy + R12 * dcz;
    float dz = R20 * dcx + R21 * dcy + R22 * dcz;
    float px = tx, py = ty, pz = tz;                 // ray_pos

    // ---- constant WMMA A-matrix: [OFFS^T ; ones ; 0...] (16x4) ----------
    float a0v = (nn == 0) ? offx(k0)     : (nn == 1) ? offy(k0)     : (nn == 2) ? offz(k0)     : (nn == 3) ? 1.0f : 0.0f;
    float a1v = (nn == 0) ? offx(k0 + 1) : (nn == 1) ? offy(k0 + 1) : (nn == 2) ? offz(k0 + 1) : (nn == 3) ? 1.0f : 0.0f;
    v2f Amat; Amat[0] = a0v; Amat[1] = a1v;

    // light = normalize([0.5, 1.0, -0.3])
    const float lrn = hrsq(0.25f + 1.0f + 0.09f);
    const float lx = 0.5f * lrn, ly = 1.0f * lrn, lz = -0.3f * lrn;

    // cos/sin of the phase offsets 2pi/3 and 4pi/3
    const float CP = -0.5f;               // cos(2pi/3) = cos(4pi/3)
    const float SP =  0.8660254037844f;   // sin(2pi/3) = -sin(4pi/3)

    float aL = 0.f, aD = 0.f, aP = 0.f, aV = 0.f;
    float aNx = 0.f, aNy = 0.f, aNz = 0.f, aLap = 0.f;
    float aT0 = 0.f, aT1 = 0.f, aT2 = 0.f;
    float aS0 = 0.f, aS1 = 0.f, aS2 = 0.f;

    for (int leg = 0; leg < legs; ++leg) {
        px += 0.1f * dx; py += 0.1f * dy; pz += 0.1f * dz;

        // sphere march (hot loop: mul, 2x fmac, v_sqrt, add, min, 3x fmac)
        for (int s = 0; s < steps; ++s) {
            float d = sdf3(px, py, pz);
            px += d * dx; py += d * dy; pz += d * dz;
        }
        float sd = sdf3(px, py, pz);

        // ---- normals + Laplacian via V_WMMA_F32_16X16X4_F32 -------------
        // Pass 0: wave-pixels [0..15]; Pass 1: wave-pixels [16..31]. In each
        // pass the lane pair (nn, nn+16) evaluates the 4 tetra samples of
        // its pixel (2 per lane) straight into the B layout.
        v8f D0, D1;
        {
            float bx = bperm_f(px, nn), by = bperm_f(py, nn), bz = bperm_f(pz, nn);
            float s0 = sdf3(bx + offx(k0),     by + offy(k0),     bz + offz(k0));
            float s1 = sdf3(bx + offx(k0 + 1), by + offy(k0 + 1), bz + offz(k0 + 1));
            v2f B; B[0] = s0; B[1] = s1;
            v8f c = {};
            D0 = __builtin_amdgcn_wmma_f32_16x16x4_f32(
                     false, Amat, false, B, (short)0, c, false, false);
        }
        {
            int src = nn + 16;
            float bx = bperm_f(px, src), by = bperm_f(py, src), bz = bperm_f(pz, src);
            float s0 = sdf3(bx + offx(k0),     by + offy(k0),     bz + offz(k0));
            float s1 = sdf3(bx + offx(k0 + 1), by + offy(k0 + 1), bz + offz(k0 + 1));
            v2f B; B[0] = s0; B[1] = s1;
            v8f c = {};
            D1 = __builtin_amdgcn_wmma_f32_16x16x4_f32(
                     false, Amat, false, B, (short)0, c, false, false);
        }
        // Branchless cross-half merge (rows 4..15 of D are zero).
        float gx = D0[0] + bperm_f(D1[0], xidx);
        float gy = D0[1] + bperm_f(D1[1], xidx);
        float gz = D0[2] + bperm_f(D1[2], xidx);
        float s4 = D0[3] + bperm_f(D1[3], xidx);

        float gin = hrcp(hsqrt(gx * gx + gy * gy + gz * gz) + 1e-12f);
        float nx_ = gx * gin, ny_ = gy * gin, nz_ = gz * gin;
        float lap = (s4 - 4.0f * sd) * (1.0f / (NEPS_C * NEPS_C));

        // ---- shading -----------------------------------------------------
        float lamb = fmaxf(0.0f, nx_ * lx + ny_ * ly + nz_ * lz);
        float mx = px - tx, my = py - ty, mz = pz - tz;
        float dist = hexp(-DECAY_C * hsqrt(mx * mx + my * my + mz * mz));
        float prox = hexp(-fabsf(sd) * (1.0f / NEPS_C));
        float vd = dx * R02 + dy * R12 + dz * R22;
        vd = fminf(fmaxf(vd, 0.0f), 1.0f);
        float vd2 = vd * vd;
        float vig = vd2 * vd2;
        float lapimg = hrcp(1.0f + hexp(-lap));
        float t1 = nx_ * R00 + ny_ * R10 + nz_ * R20;
        float t2 = nx_ * R01 + ny_ * R11 + nz_ * R21;
        float ang = atan2f(t2, t1);

        // One sincos for all 6 cosines:
        //   tang_c = cos(a + phi_c), phi = {0, 2pi/3, 4pi/3}
        //   spin_c = cos(2a + phi_c)  (2a exactly; double-angle identities)
        float a1 = degf * ang;
        float sa, ca;
        sincosf(a1, &sa, &ca);
        float c2 = 2.0f * ca * ca - 1.0f;       // cos(2*a1)
        float s2 = 2.0f * sa * ca;              // sin(2*a1)
        float tg0 = ca;
        float tg1 = CP * ca - SP * sa;          // cos(a1 + 2pi/3)
        float tg2 = CP * ca + SP * sa;          // cos(a1 + 4pi/3)
        float sp0 = c2;
        float sp1 = CP * c2 - SP * s2;
        float sp2 = CP * c2 + SP * s2;

        aL += lamb; aD += dist; aP += prox; aV += vig;
        aNx += nx_ * 0.5f + 0.5f; aNy += ny_ * 0.5f + 0.5f; aNz += nz_ * 0.5f + 0.5f;
        aLap += lapimg;
        aT0 += 0.5f + 0.5f * tg0;
        aT1 += 0.5f + 0.5f * tg1;
        aT2 += 0.5f + 0.5f * tg2;
        aS0 += 0.5f + 0.5f * sp0;
        aS1 += 0.5f + 0.5f * sp1;
        aS2 += 0.5f + 0.5f * sp2;

        // ---- reflect + advance ------------------------------------------
        float pd = -(nx_ * dx + ny_ * dy + nz_ * dz);
        dx += 2.0f * pd * nx_;
        dy += 2.0f * pd * ny_;
        dz += 2.0f * pd * nz_;
        // ray_pos = marched (px,py,pz already hold it)
    }

    // ---- write 8 images (scalars broadcast to 3 channels) ----------------
    const float invL = hrcp((float)legs);   // exact for small powers of two
    const size_t HW3 = (size_t)PXN * PXN * 3;
    const size_t p3  = (size_t)gid * 3;

    float* o0 = out + 0 * HW3 + p3;   // lamb
    float* o1 = out + 1 * HW3 + p3;   // dist
    float* o2 = out + 2 * HW3 + p3;   // prox
    float* o3 = out + 3 * HW3 + p3;   // vig
    float* o4 = out + 4 * HW3 + p3;   // nrm
    float* o5 = out + 5 * HW3 + p3;   // lap_img
    float* o6 = out + 6 * HW3 + p3;   // tang
    float* o7 = out + 7 * HW3 + p3;   // spin

    float vL = aL * invL, vD = aD * invL, vP = aP * invL, vV = aV * invL, vLap = aLap * invL;
    o0[0] = vL;  o0[1] = vL;  o0[2] = vL;
    o1[0] = vD;  o1[1] = vD;  o1[2] = vD;
    o2[0] = vP;  o2[1] = vP;  o2[2] = vP;
    o3[0] = vV;  o3[1] = vV;  o3[2] = vV;
    o4[0] = aNx * invL; o4[1] = aNy * invL; o4[2] = aNz * invL;
    o5[0] = vLap; o5[1] = vLap; o5[2] = vLap;
    o6[0] = aT0 * invL; o6[1] = aT1 * invL; o6[2] = aT2 * invL;
    o7[0] = aS0 * invL; o7[1] = aS1 * invL; o7[2] = aS2 * invL;
}

extern "C" void kernel_launch(void* const* d_in, const int* in_sizes, int n_in,
                              void* d_out, int out_size, void* d_ws, size_t ws_size,
                              hipStream_t stream) {
    (void)in_sizes; (void)n_in; (void)out_size; (void)d_ws; (void)ws_size;
    const float* q      = (const float*)d_in[0];   // orientations (1,4) f32
    const float* t      = (const float*)d_in[1];   // translations (1,3) f32
    const int*   degree = (const int*)d_in[2];
    const int*   steps  = (const int*)d_in[3];
    const int*   legs   = (const int*)d_in[4];
    float* out = (float*)d_out;

    dim3 block(256);                               // 8 wave32s
    dim3 grid((PXN * PXN) / 256);                  // exact: 2500 blocks
    render_raymarch<<<grid, block, 0, stream>>>(q, t, degree, steps, legs, out);
}